// BRASKModel_11398843203746
// MI455X (gfx1250) — compile-verified
//
#include <hip/hip_runtime.h>
#include <math.h>

// ---------------------------------------------------------------------------
// Shapes (fixed): B=4, L=256, R=64, H=768, TE=100
// ---------------------------------------------------------------------------
#define H    768
#define H2   (768 * 768)
#define L    256
#define Bsz  4
#define R    64
#define TE   100

typedef float v2f __attribute__((ext_vector_type(2)));
typedef float v8f __attribute__((ext_vector_type(8)));

// d_out layout (floats), tuple return order:
#define OFF_FSS 0          // f_sub_s [4,256]
#define OFF_FSE 1024       // f_sub_e
#define OFF_FOS 2048       // f_obj_s [4,256,64]
#define OFF_FOE 67584      // f_obj_e
#define OFF_BOS 133120     // b_obj_s [4,256]
#define OFF_BOE 134144     // b_obj_e
#define OFF_BSS 135168     // b_sub_s [4,256,64]
#define OFF_BSE 200704     // b_sub_e

// workspace layout (float offsets)
#define WS_CSUM    0                       // [4][257][768]
#define WS_FPAD    (WS_CSUM + 4*257*768)   // [1024][768]
#define WS_BPAD    (WS_FPAD + 1024*768)
#define WS_G0      (WS_BPAD + 1024*768)    // f_pad@Wm0+Wb0
#define WS_G1      (WS_G0   + 1024*768)
#define WS_TOKF    (WS_G1   + 1024*768)    // embs@Wm4+Wb4
#define WS_TOKB    (WS_TOKF + 1024*768)
#define WS_G8      (WS_TOKB + 1024*768)
#define WS_G9      (WS_G8   + 1024*768)
#define WS_CF      (WS_G9   + 1024*768)    // [4][64][768]
#define WS_CB      (WS_CF   + 4*64*768)
#define WS_RELPF   (WS_CB   + 4*64*768)    // [64][768]
#define WS_RELPB   (WS_RELPF + 64*768)
#define WS_BREL    (WS_RELPB + 64*768)
#define WS_HGPF    (WS_BREL + 64*768)      // [4][768]
#define WS_HGPB    (WS_HGPF + 4*768)
#define WS_TF4     (WS_HGPB + 4*768)       // [1024] each
#define WS_TF5     (WS_TF4 + 1024)
#define WS_TB6     (WS_TF5 + 1024)
#define WS_TB7     (WS_TB6 + 1024)
#define WS_CF4     (WS_TB7 + 1024)         // [256] each
#define WS_CF5     (WS_CF4 + 256)
#define WS_CB6     (WS_CF5 + 256)
#define WS_CB7     (WS_CB6 + 256)
#define WS_SLIF    (WS_CB7 + 256)          // int [1024] each
#define WS_SLJF    (WS_SLIF + 1024)
#define WS_SLIB    (WS_SLJF + 1024)
#define WS_SLJB    (WS_SLIB + 1024)
#define WS_MSKF    (WS_SLJB + 1024)        // float [1024]
#define WS_MSKB    (WS_MSKF + 1024)

// ---------------------------------------------------------------------------
// WMMA f32 GEMM:  Y[M x 768] = X[M x 768] @ W[768 x 768] + bias[768]
// grid = (768/64, M/64), block = 256 (8 waves, 4(M) x 2(N) wave grid).
// Each wave owns a 16x32 tile: two v8f accumulators sharing one A fragment,
// so DS traffic is 3 dwords per v_wmma instead of 4. K staged via LDS (KC=32),
// next K-chunk prefetched (global_prefetch_b8) while WMMAs run.
// ---------------------------------------------------------------------------
#define KC 32
__global__ __launch_bounds__(256) void gemm_wmma_f32(
    const float* __restrict__ X, const float* __restrict__ W,
    const float* __restrict__ bias, float* __restrict__ Y)
{
  __shared__ float As[KC][64 + 1];   // [k][m], padded vs bank conflicts
  __shared__ float Bs[KC][64 + 1];   // [k][n]
  const int tid   = threadIdx.x;
  const int lane  = tid & 31;
  const int wave  = tid >> 5;
  const int waveM = wave & 3;        // 0..3  -> rows  waveM*16
  const int waveN = wave >> 2;       // 0..1  -> cols  waveN*32
  const int rowBase = blockIdx.y * 64;
  const int colBase = blockIdx.x * 64;
  const int hlf = lane >> 4;         // 0: K pair {0,1}; 1: K pair {2,3}
  const int l16 = lane & 15;
  const int lm = waveM * 16 + l16;
  const int ln = waveN * 32 + l16;

  v8f acc0 = {};
  v8f acc1 = {};
  for (int k0 = 0; k0 < H; k0 += KC) {
    // A tile: 64 rows x 32 K (coalesced along K)
    #pragma unroll
    for (int t = 0; t < 8; ++t) {
      int idx = tid + t * 256;           // 0..2047
      int m = idx >> 5, k = idx & 31;
      As[k][m] = X[(rowBase + m) * H + k0 + k];
    }
    // B tile: 32 K x 64 N (coalesced along N)
    #pragma unroll
    for (int t = 0; t < 8; ++t) {
      int idx = tid + t * 256;           // 0..2047
      int k = idx >> 6, n = idx & 63;
      Bs[k][n] = W[(k0 + k) * H + colBase + n];
    }
    // pull next K-chunk toward the WGP while this chunk computes
    if (k0 + KC < H) {
      __builtin_prefetch(&X[(rowBase + (tid >> 5)) * H + k0 + KC + (tid & 31)], 0, 1);
      __builtin_prefetch(&W[(k0 + KC + (tid >> 6)) * H + colBase + (tid & 63)], 0, 1);
    }
    __syncthreads();
    #pragma unroll
    for (int kk = 0; kk < KC; kk += 4) {
      // ISA 7.12.2: lanes 0-15 hold (K=kk, K=kk+1), lanes 16-31 (K=kk+2, kk+3)
      v2f a  = { As[kk + 2 * hlf + 0][lm],      As[kk + 2 * hlf + 1][lm] };
      v2f b0 = { Bs[kk + 2 * hlf + 0][ln],      Bs[kk + 2 * hlf + 1][ln] };
      v2f b1 = { Bs[kk + 2 * hlf + 0][ln + 16], Bs[kk + 2 * hlf + 1][ln + 16] };
      acc0 = __builtin_amdgcn_wmma_f32_16x16x4_f32(
          false, a, false, b0, (short)0, acc0, false, false);
      acc1 = __builtin_amdgcn_wmma_f32_16x16x4_f32(
          false, a, false, b1, (short)0, acc1, false, false);
    }
    __syncthreads();
  }
  // C/D layout: VGPR i -> lanes 0-15: M=i, lanes 16-31: M=i+8; N = lane%16
  const int n0 = colBase + waveN * 32 + l16;
  const int n1 = n0 + 16;
  const float bv0 = bias[n0];
  const float bv1 = bias[n1];
  #pragma unroll
  for (int i = 0; i < 8; ++i) {
    int m = rowBase + waveM * 16 + i + 8 * hlf;
    Y[m * H + n0] = acc0[i] + bv0;
    Y[m * H + n1] = acc1[i] + bv1;
  }
}

// ---------------------------------------------------------------------------
// b_rel[r,h] = rel_transe[r,:] @ rproj_w[:,h] + rproj_b[h]   (K=100)
// ---------------------------------------------------------------------------
__global__ __launch_bounds__(256) void brel_kernel(
    const float* __restrict__ rte, const float* __restrict__ rpw,
    const float* __restrict__ rpb, float* __restrict__ brel)
{
  const int r = blockIdx.x;
  for (int h = threadIdx.x; h < H; h += 256) {
    float s = rpb[h];
    for (int k = 0; k < TE; ++k) s += rte[r * TE + k] * rpw[k * H + h];
    brel[r * H + h] = s;
  }
}

// hgp[b,h] = h_gs[b,:] @ Wm_i[:,h] + Wb_i[h]
__global__ __launch_bounds__(256) void hgp_kernel(
    const float* __restrict__ hg, const float* __restrict__ Wmi,
    const float* __restrict__ Wbi, float* __restrict__ outp)
{
  const int b = blockIdx.x;
  for (int h = threadIdx.x; h < H; h += 256) {
    float s = Wbi[h];
    for (int k = 0; k < H; ++k) s += hg[b * H + k] * Wmi[k * H + h];
    outp[b * H + h] = s;
  }
}

// ---------------------------------------------------------------------------
// token logits: 4 dots per token vs fc_w rows 0..3. One wave per token.
// ---------------------------------------------------------------------------
__global__ __launch_bounds__(256) void tok_logits_kernel(
    const float* __restrict__ embs, const float* __restrict__ fc_w,
    const float* __restrict__ fc_b, float* __restrict__ out)
{
  const int tok = blockIdx.x * 8 + (threadIdx.x >> 5);
  const int lane = threadIdx.x & 31;
  const float* x = embs + tok * H;
  float a0 = 0, a1 = 0, a2 = 0, a3 = 0;
  for (int h = lane; h < H; h += 32) {
    float xv = x[h];
    a0 += xv * fc_w[0 * H + h];
    a1 += xv * fc_w[1 * H + h];
    a2 += xv * fc_w[2 * H + h];
    a3 += xv * fc_w[3 * H + h];
  }
  #pragma unroll
  for (int off = 16; off > 0; off >>= 1) {
    a0 += __shfl_down(a0, off, 32);
    a1 += __shfl_down(a1, off, 32);
    a2 += __shfl_down(a2, off, 32);
    a3 += __shfl_down(a3, off, 32);
  }
  if (lane == 0) {
    out[OFF_FSS + tok] = a0 + fc_b[0];
    out[OFF_FSE + tok] = a1 + fc_b[1];
    out[OFF_BOS + tok] = a2 + fc_b[2];
    out[OFF_BOE + tok] = a3 + fc_b[3];
  }
}

// per-(b,h) prefix sums of embs along L: csum[b][0..256][h]
__global__ __launch_bounds__(256) void csum_kernel(
    const float* __restrict__ embs, float* __restrict__ csum)
{
  const int idx = blockIdx.x * 256 + threadIdx.x;   // 0..3071
  const int b = idx / H, h = idx % H;
  float s = 0.0f;
  csum[(b * 257 + 0) * H + h] = 0.0f;
  for (int l = 0; l < L; ++l) {
    s += embs[(b * L + l) * H + h];
    csum[(b * 257 + l + 1) * H + h] = s;
  }
}

// ---------------------------------------------------------------------------
// span control: suffix-min of end positions, validity, rank prefix-sum,
// slot scatter + mask. One 256-thread block per batch.
// sigmoid(x) > 0.5  <=>  x > 0.
// ---------------------------------------------------------------------------
__global__ __launch_bounds__(256) void span_ctl_kernel(
    const float* __restrict__ s_log, const float* __restrict__ e_log,
    int* __restrict__ slotI, int* __restrict__ slotJ, float* __restrict__ mask)
{
  const int b = blockIdx.x;
  const int l = threadIdx.x;
  __shared__ int ne[256];
  __shared__ int rk[256];
  const bool start = s_log[b * L + l] > 0.0f;
  const bool end   = e_log[b * L + l] > 0.0f;
  ne[l] = end ? l : L;
  __syncthreads();
  for (int off = 1; off < L; off <<= 1) {       // suffix min (reverse cummin)
    int v = ne[l];
    int o = (l + off < L) ? ne[l + off] : L;
    __syncthreads();
    ne[l] = min(v, o);
    __syncthreads();
  }
  const int nxt = ne[l];
  const bool valid = start && (nxt < L);
  const int j = min(nxt, L - 1);
  rk[l] = valid ? 1 : 0;
  __syncthreads();
  for (int off = 1; off < L; off <<= 1) {       // inclusive prefix sum
    int v = rk[l];
    int o = (l >= off) ? rk[l - off] : 0;
    __syncthreads();
    rk[l] = v + o;
    __syncthreads();
  }
  const int num = rk[L - 1];
  mask[b * L + l] = (l < num) ? 1.0f : 0.0f;
  if (valid) {
    int rank = rk[l] - 1;
    slotI[b * L + rank] = l;
    slotJ[b * L + rank] = j;
  }
}

// pad[b,s,:] = (csum[b,j+1,:] - csum[b,i,:]) / (j-i+1) for valid slots, else 0
__global__ __launch_bounds__(256) void pad_build_kernel(
    const float* __restrict__ csum, const int* __restrict__ slotI,
    const int* __restrict__ slotJ, const float* __restrict__ mask,
    float* __restrict__ pad)
{
  const int b = blockIdx.x >> 8;
  const int s = blockIdx.x & 255;
  const float mk = mask[b * L + s];
  int i = 0, j = 0;
  float inv = 0.0f;
  if (mk > 0.5f) {
    i = slotI[b * L + s];
    j = slotJ[b * L + s];
    inv = 1.0f / (float)(j - i + 1);
  }
  for (int h = threadIdx.x; h < H; h += 256) {
    float v = 0.0f;
    if (mk > 0.5f)
      v = (csum[(b * 257 + j + 1) * H + h] - csum[(b * 257 + i) * H + h]) * inv;
    pad[(b * L + s) * H + h] = v;
  }
}

// ---------------------------------------------------------------------------
// fused attention per (b,r): v[l] = V_b + sum_h V_w[h]*tanh(tokp+relp+hgp);
// softmax over l; C[b,r,h] = sum_l A[l]*embs[b,l,h].
// One 256-thread block (8 waves) per (b,r).
// ---------------------------------------------------------------------------
__global__ __launch_bounds__(256) void attn_kernel(
    const float* __restrict__ tokp, const float* __restrict__ relp,
    const float* __restrict__ hgp, const float* __restrict__ embs,
    const float* __restrict__ V_w, const float* __restrict__ V_b,
    float* __restrict__ C)
{
  const int b = blockIdx.x >> 6;
  const int r = blockIdx.x & 63;
  __shared__ float sh_rh[H];
  __shared__ float sh_v[L];
  __shared__ float sh_red[8];
  const int tid = threadIdx.x;
  const int lane = tid & 31;
  const int wave = tid >> 5;

  for (int h = tid; h < H; h += 256)
    sh_rh[h] = relp[r * H + h] + hgp[b * H + h];
  __syncthreads();

  const float vb = V_b[0];
  for (int l = wave; l < L; l += 8) {
    const float* tp = tokp + (b * L + l) * H;
    float s = 0.0f;
    for (int h = lane; h < H; h += 32)
      s += V_w[h] * tanhf(tp[h] + sh_rh[h]);
    #pragma unroll
    for (int off = 16; off > 0; off >>= 1) s += __shfl_down(s, off, 32);
    if (lane == 0) sh_v[l] = s + vb;
  }
  __syncthreads();

  // block softmax over L
  const float x = sh_v[tid];
  float wm = x;
  #pragma unroll
  for (int off = 16; off > 0; off >>= 1) wm = fmaxf(wm, __shfl_down(wm, off, 32));
  if (lane == 0) sh_red[wave] = wm;
  __syncthreads();
  float m = sh_red[0];
  #pragma unroll
  for (int i = 1; i < 8; ++i) m = fmaxf(m, sh_red[i]);
  const float e = expf(x - m);
  __syncthreads();
  float ps = e;
  #pragma unroll
  for (int off = 16; off > 0; off >>= 1) ps += __shfl_down(ps, off, 32);
  if (lane == 0) sh_red[wave] = ps;
  __syncthreads();
  float tot = 0.0f;
  #pragma unroll
  for (int i = 0; i < 8; ++i) tot += sh_red[i];
  sh_v[tid] = e / tot;
  __syncthreads();

  for (int h = tid; h < H; h += 256) {
    float acc = 0.0f;
    for (int l = 0; l < L; ++l)
      acc += sh_v[l] * embs[(b * L + l) * H + h];
    C[(b * R + r) * H + h] = acc;
  }
}

// ---------------------------------------------------------------------------
// t dots: t_i[b,l] = (G*mask + Gtok + embs) . fc_w[i]; one wave per token
// ---------------------------------------------------------------------------
__global__ __launch_bounds__(256) void tdots_kernel(
    const float* __restrict__ G0, const float* __restrict__ G8,
    const float* __restrict__ G1, const float* __restrict__ G9,
    const float* __restrict__ embs, const float* __restrict__ maskf,
    const float* __restrict__ maskb, const float* __restrict__ fc_w,
    float* __restrict__ t_f4, float* __restrict__ t_f5,
    float* __restrict__ t_b6, float* __restrict__ t_b7)
{
  const int tok = blockIdx.x * 8 + (threadIdx.x >> 5);
  const int lane = threadIdx.x & 31;
  const float mf = maskf[tok], mb = maskb[tok];
  const float* e  = embs + tok * H;
  const float* g0 = G0 + tok * H;
  const float* g8 = G8 + tok * H;
  const float* g1 = G1 + tok * H;
  const float* g9 = G9 + tok * H;
  float a4 = 0, a5 = 0, a6 = 0, a7 = 0;
  for (int h = lane; h < H; h += 32) {
    float ev = e[h];
    float hf = g0[h] * mf + g8[h] + ev;
    float hb = g1[h] * mb + g9[h] + ev;
    a4 += hf * fc_w[4 * H + h];
    a5 += hf * fc_w[5 * H + h];
    a6 += hb * fc_w[6 * H + h];
    a7 += hb * fc_w[7 * H + h];
  }
  #pragma unroll
  for (int off = 16; off > 0; off >>= 1) {
    a4 += __shfl_down(a4, off, 32);
    a5 += __shfl_down(a5, off, 32);
    a6 += __shfl_down(a6, off, 32);
    a7 += __shfl_down(a7, off, 32);
  }
  if (lane == 0) { t_f4[tok] = a4; t_f5[tok] = a5; t_b6[tok] = a6; t_b7[tok] = a7; }
}

// c dots: c_i[b,r] = C[b,r,:] . fc_w[i]; one wave per (b,r)
__global__ __launch_bounds__(256) void cdots_kernel(
    const float* __restrict__ C_f, const float* __restrict__ C_b,
    const float* __restrict__ fc_w, float* __restrict__ c_f4,
    float* __restrict__ c_f5, float* __restrict__ c_b6, float* __restrict__ c_b7)
{
  const int pr = blockIdx.x * 8 + (threadIdx.x >> 5);   // 0..255
  const int lane = threadIdx.x & 31;
  const float* cf = C_f + pr * H;
  const float* cb = C_b + pr * H;
  float a4 = 0, a5 = 0, a6 = 0, a7 = 0;
  for (int h = lane; h < H; h += 32) {
    a4 += cf[h] * fc_w[4 * H + h];
    a5 += cf[h] * fc_w[5 * H + h];
    a6 += cb[h] * fc_w[6 * H + h];
    a7 += cb[h] * fc_w[7 * H + h];
  }
  #pragma unroll
  for (int off = 16; off > 0; off >>= 1) {
    a4 += __shfl_down(a4, off, 32);
    a5 += __shfl_down(a5, off, 32);
    a6 += __shfl_down(a6, off, 32);
    a7 += __shfl_down(a7, off, 32);
  }
  if (lane == 0) { c_f4[pr] = a4; c_f5[pr] = a5; c_b6[pr] = a6; c_b7[pr] = a7; }
}

// final pair logits: out[b,l,r] = t[b,l] + c[b,r] + fc_b[i] for the 4 outputs
__global__ __launch_bounds__(256) void final_kernel(
    const float* __restrict__ t_f4, const float* __restrict__ t_f5,
    const float* __restrict__ t_b6, const float* __restrict__ t_b7,
    const float* __restrict__ c_f4, const float* __restrict__ c_f5,
    const float* __restrict__ c_b6, const float* __restrict__ c_b7,
    const float* __restrict__ fc_b, float* __restrict__ out)
{
  const int idx = blockIdx.x * 256 + threadIdx.x;   // 0..65535
  const int b = idx >> 14;
  const int l = (idx >> 6) & 255;
  const int r = idx & 63;
  const int tok = b * L + l;
  const int pr  = b * R + r;
  out[OFF_FOS + idx] = t_f4[tok] + c_f4[pr] + fc_b[4];
  out[OFF_FOE + idx] = t_f5[tok] + c_f5[pr] + fc_b[5];
  out[OFF_BSS + idx] = t_b6[tok] + c_b6[pr] + fc_b[6];
  out[OFF_BSE + idx] = t_b7[tok] + c_b7[pr] + fc_b[7];
}

// ---------------------------------------------------------------------------
extern "C" void kernel_launch(void* const* d_in, const int* in_sizes, int n_in,
                              void* d_out, int out_size, void* d_ws, size_t ws_size,
                              hipStream_t stream)
{
  (void)in_sizes; (void)n_in; (void)out_size; (void)ws_size;
  const float* embs     = (const float*)d_in[0];
  const float* h_gs     = (const float*)d_in[1];
  const float* rel_embs = (const float*)d_in[2];
  const float* rte      = (const float*)d_in[3];
  const float* fc_w     = (const float*)d_in[4];
  const float* fc_b     = (const float*)d_in[5];
  const float* Wm       = (const float*)d_in[6];
  const float* Wb       = (const float*)d_in[7];
  const float* V_w      = (const float*)d_in[8];
  const float* V_b      = (const float*)d_in[9];
  const float* rproj_w  = (const float*)d_in[10];
  const float* rproj_b  = (const float*)d_in[11];
  float* out = (float*)d_out;
  float* ws  = (float*)d_ws;

  float* csum   = ws + WS_CSUM;
  float* f_pad  = ws + WS_FPAD;
  float* b_pad  = ws + WS_BPAD;
  float* G0     = ws + WS_G0;
  float* G1     = ws + WS_G1;
  float* tokf   = ws + WS_TOKF;
  float* tokb   = ws + WS_TOKB;
  float* G8     = ws + WS_G8;
  float* G9     = ws + WS_G9;
  float* C_f    = ws + WS_CF;
  float* C_b    = ws + WS_CB;
  float* relp_f = ws + WS_RELPF;
  float* relp_b = ws + WS_RELPB;
  float* brel   = ws + WS_BREL;
  float* hgp_f  = ws + WS_HGPF;
  float* hgp_b  = ws + WS_HGPB;
  float* t_f4   = ws + WS_TF4;
  float* t_f5   = ws + WS_TF5;
  float* t_b6   = ws + WS_TB6;
  float* t_b7   = ws + WS_TB7;
  float* c_f4   = ws + WS_CF4;
  float* c_f5   = ws + WS_CF5;
  float* c_b6   = ws + WS_CB6;
  float* c_b7   = ws + WS_CB7;
  int* slotI_f  = (int*)(ws + WS_SLIF);
  int* slotJ_f  = (int*)(ws + WS_SLJF);
  int* slotI_b  = (int*)(ws + WS_SLIB);
  int* slotJ_b  = (int*)(ws + WS_SLJB);
  float* mask_f = ws + WS_MSKF;
  float* mask_b = ws + WS_MSKB;

  // small projections
  brel_kernel<<<R, 256, 0, stream>>>(rte, rproj_w, rproj_b, brel);
  hgp_kernel<<<Bsz, 256, 0, stream>>>(h_gs, Wm + 3 * H2, Wb + 3 * H, hgp_f);
  hgp_kernel<<<Bsz, 256, 0, stream>>>(h_gs, Wm + 6 * H2, Wb + 6 * H, hgp_b);
  gemm_wmma_f32<<<dim3(H / 64, R / 64), 256, 0, stream>>>(rel_embs, Wm + 2 * H2, Wb + 2 * H, relp_f);
  gemm_wmma_f32<<<dim3(H / 64, R / 64), 256, 0, stream>>>(brel,     Wm + 5 * H2, Wb + 5 * H, relp_b);

  // token logits + span extraction
  tok_logits_kernel<<<128, 256, 0, stream>>>(embs, fc_w, fc_b, out);
  csum_kernel<<<(Bsz * H) / 256, 256, 0, stream>>>(embs, csum);
  span_ctl_kernel<<<Bsz, 256, 0, stream>>>(out + OFF_FSS, out + OFF_FSE, slotI_f, slotJ_f, mask_f);
  span_ctl_kernel<<<Bsz, 256, 0, stream>>>(out + OFF_BOS, out + OFF_BOE, slotI_b, slotJ_b, mask_b);
  pad_build_kernel<<<Bsz * L, 256, 0, stream>>>(csum, slotI_f, slotJ_f, mask_f, f_pad);
  pad_build_kernel<<<Bsz * L, 256, 0, stream>>>(csum, slotI_b, slotJ_b, mask_b, b_pad);

  // big WMMA GEMMs: [1024x768] @ [768x768] + bias
  dim3 gBig(H / 64, (Bsz * L) / 64);
  gemm_wmma_f32<<<gBig, 256, 0, stream>>>(f_pad, Wm + 0 * H2, Wb + 0 * H, G0);
  gemm_wmma_f32<<<gBig, 256, 0, stream>>>(b_pad, Wm + 1 * H2, Wb + 1 * H, G1);
  gemm_wmma_f32<<<gBig, 256, 0, stream>>>(embs,  Wm + 4 * H2, Wb + 4 * H, tokf);
  gemm_wmma_f32<<<gBig, 256, 0, stream>>>(embs,  Wm + 7 * H2, Wb + 7 * H, tokb);
  gemm_wmma_f32<<<gBig, 256, 0, stream>>>(embs,  Wm + 8 * H2, Wb + 8 * H, G8);
  gemm_wmma_f32<<<gBig, 256, 0, stream>>>(embs,  Wm + 9 * H2, Wb + 9 * H, G9);

  // fused attention (the VALU/tanh-heavy part)
  attn_kernel<<<Bsz * R, 256, 0, stream>>>(tokf, relp_f, hgp_f, embs, V_w, V_b, C_f);
  attn_kernel<<<Bsz * R, 256, 0, stream>>>(tokb, relp_b, hgp_b, embs, V_w, V_b, C_b);

  // pair logits
  tdots_kernel<<<128, 256, 0, stream>>>(G0, G8, G1, G9, embs, mask_f, mask_b, fc_w,
                                        t_f4, t_f5, t_b6, t_b7);
  cdots_kernel<<<32, 256, 0, stream>>>(C_f, C_b, fc_w, c_f4, c_f5, c_b6, c_b7);
  final_kernel<<<(Bsz * L * R) / 256, 256, 0, stream>>>(t_f4, t_f5, t_b6, t_b7,
                                                        c_f4, c_f5, c_b6, c_b7, fc_b, out);
}